// SumPooling_39127152066611
// MI455X (gfx1250) — compile-verified
//
#include <hip/hip_runtime.h>
#include <hip/hip_bf16.h>
#include <stdint.h>

#define BLOCK 256
#define DIM 32

#if defined(__has_builtin)
#if __has_builtin(__builtin_amdgcn_global_load_async_to_lds_b128)
#define HAVE_ASYNC_LDS 1
#endif
#endif

// 16-byte vector type matching the builtin's expected pointee
typedef int v4i __attribute__((ext_vector_type(4)));
typedef __attribute__((address_space(1))) v4i* gv4i_ptr;   // global (AS1)
typedef __attribute__((address_space(3))) v4i* lv4i_ptr;   // LDS (AS3)

__global__ void zero_kernel(float* __restrict__ out, int n) {
  int i = blockIdx.x * blockDim.x + threadIdx.x;
  if (i < n) out[i] = 0.0f;
}

__global__ void __launch_bounds__(BLOCK)
seg_sum_kernel(const float* __restrict__ x,
               const int* __restrict__ idx,
               float* __restrict__ out,
               int n) {
  __shared__ float tile[BLOCK * DIM];   // 32 KB of the 320 KB WGP LDS
  __shared__ int sidx[BLOCK];

  const int tid = threadIdx.x;
  const long long base = (long long)blockIdx.x * BLOCK;  // first row of tile

  if (base + BLOCK <= (long long)n) {
    // ---- fast path: full 256-row tile ----
    const float* gsrc = x + base * DIM;

#if defined(HAVE_ASYNC_LDS)
    // Async copy engine: 8 x b128 per thread -> 32 KB tile, no VGPR returns.
#pragma unroll
    for (int j = 0; j < 8; ++j) {
      int off = (j * BLOCK + tid) * 4;  // float index, 16B per lane per issue
      __builtin_amdgcn_global_load_async_to_lds_b128(
          (gv4i_ptr)(gsrc + off), (lv4i_ptr)(&tile[off]),
          /*offset=*/0, /*cpol=*/0);
    }
    sidx[tid] = idx[base + tid];        // overlap with async copies
    asm volatile("s_wait_asynccnt 0" ::: "memory");
#else
    // Fallback: coalesced b128 loads through VGPRs
    const float4* gsrc4 = reinterpret_cast<const float4*>(gsrc);
    float4* tile4w = reinterpret_cast<float4*>(tile);
#pragma unroll
    for (int j = 0; j < 8; ++j) {
      int off = j * BLOCK + tid;        // float4 index
      tile4w[off] = gsrc4[off];
    }
    sidx[tid] = idx[base + tid];
#endif
    __syncthreads();

    // ---- drain: LDS b128 reads -> 4 fire-and-forget fp32 atomics each ----
    const float4* tile4 = reinterpret_cast<const float4*>(tile);
#pragma unroll
    for (int k = 0; k < 8; ++k) {
      int e4 = k * BLOCK + tid;         // float4 slot
      int e = e4 * 4;                   // float index within tile
      float4 v = tile4[e4];
      int row = e >> 5;                 // DIM == 32
      int col = e & 31;                 // multiple of 4, so row is uniform per quad
      long long o = (long long)sidx[row] * DIM + col;
      unsafeAtomicAdd(&out[o + 0], v.x);
      unsafeAtomicAdd(&out[o + 1], v.y);
      unsafeAtomicAdd(&out[o + 2], v.z);
      unsafeAtomicAdd(&out[o + 3], v.w);
    }
  } else {
    // ---- tail path: guarded, one row per thread ----
    for (long long r = base + tid; r < (long long)n; r += BLOCK) {
      int seg = idx[r];
      long long obase = (long long)seg * DIM;
      long long ibase = r * DIM;
#pragma unroll
      for (int c = 0; c < DIM; ++c) {
        unsafeAtomicAdd(&out[obase + c], x[ibase + c]);
      }
    }
  }
}

extern "C" void kernel_launch(void* const* d_in, const int* in_sizes, int n_in,
                              void* d_out, int out_size, void* d_ws, size_t ws_size,
                              hipStream_t stream) {
  const float* x = (const float*)d_in[0];
  const int* idx = (const int*)d_in[1];   // dst_idx (int32 per harness convention)
  float* out = (float*)d_out;

  const int n = in_sizes[1];              // N_SRC rows (in_sizes[0] == n * 32)

  // 1) zero the accumulator (d_out is poisoned before timing)
  int zblocks = (out_size + BLOCK - 1) / BLOCK;
  zero_kernel<<<zblocks, BLOCK, 0, stream>>>(out, out_size);

  // 2) scatter-add
  int tiles = (n + BLOCK - 1) / BLOCK;
  seg_sum_kernel<<<tiles, BLOCK, 0, stream>>>(x, idx, out, n);
}